// development_layer_56470230008339
// MI455X (gfx1250) — compile-verified
//
#include <hip/hip_runtime.h>

// Path-development layer for MI455X (gfx1250, wave32).
//   x: (64, 2048, 8) f32, A: (2, 8, 10, 10) f32  ->  out: (64, 2, 10, 10) f32
// One wave owns one padded 16x16 f32 tile; all matmuls are chains of four
// v_wmma_f32_16x16x4_f32. expm via scaling-squaring (s=1) + degree-8 Taylor
// Horner. 4096 independent chunk-chains, then an ordered sequential fold.

typedef __attribute__((ext_vector_type(2))) float v2f;
typedef __attribute__((ext_vector_type(8))) float v8f;

#define N_      64
#define T_      2048
#define IN_     8
#define C_      2
#define M_      10
#define NT_     (T_ - 1)   // 2047 increments
#define CHUNKS  32
#define CS_     64         // ceil(2047/32)

// 16x16x16 f32 matmul: acc += A(lsA, row-major 16x16) @ B(lsB, row-major 16x16)
__device__ __attribute__((always_inline)) inline
v8f mm16_acc(const float* lsA, const float* lsB, v8f acc, int lo, int hi) {
#pragma unroll
  for (int kk = 0; kk < 4; ++kk) {
    v2f a, b;
    a.x = lsA[lo * 16 + (kk * 4 + 2 * hi + 0)];
    a.y = lsA[lo * 16 + (kk * 4 + 2 * hi + 1)];
    b.x = lsB[(kk * 4 + 2 * hi + 0) * 16 + lo];
    b.y = lsB[(kk * 4 + 2 * hi + 1) * 16 + lo];
    acc = __builtin_amdgcn_wmma_f32_16x16x4_f32(false, a, false, b,
                                                (short)0, acc, false, false);
  }
  return acc;
}

__global__ __launch_bounds__(32)
void dev_chunk_kernel(const float* __restrict__ x, const float* __restrict__ A,
                      float* __restrict__ ws) {
  __shared__ float lsA[256];
  __shared__ float lsB[256];

  const int lane = threadIdx.x;
  const int lo = lane & 15;
  const int hi = lane >> 4;

  const int wid = blockIdx.x;          // 0 .. N*C*CHUNKS-1
  const int q   = wid % CHUNKS;
  const int nc  = wid / CHUNKS;
  const int c   = nc % C_;
  const int n   = nc / C_;

  // Preload skew generators into C-fragment registers, pre-scaled by 0.5
  // (scaling-squaring s=1): sk[i][v] = 0.5*(A[c,i,r,co] - A[c,i,co,r])
  float sk[IN_][8];
#pragma unroll
  for (int v = 0; v < 8; ++v) {
    const int r = v + 8 * hi;
#pragma unroll
    for (int i = 0; i < IN_; ++i) {
      float val = 0.0f;
      if (r < M_ && lo < M_) {
        const float* Ai = A + ((size_t)(c * IN_ + i)) * (M_ * M_);
        val = 0.5f * (Ai[r * M_ + lo] - Ai[lo * M_ + r]);
      }
      sk[i][v] = val;
    }
  }

  // Running product P = I (C-fragment)
  v8f P;
#pragma unroll
  for (int v = 0; v < 8; ++v) P[v] = ((v + 8 * hi) == lo) ? 1.0f : 0.0f;

  const int t0 = q * CS_;
  const int t1 = (t0 + CS_ < NT_) ? (t0 + CS_) : NT_;

  const float* xr = x + (size_t)n * T_ * IN_;
  float xc[IN_];
#pragma unroll
  for (int i = 0; i < IN_; ++i) xc[i] = xr[t0 * IN_ + i];

  for (int t = t0; t < t1; ++t) {
    float xn[IN_];
#pragma unroll
    for (int i = 0; i < IN_; ++i) xn[i] = xr[(t + 1) * IN_ + i];

    // Gh = 0.5 * G  (C-fragment), 64 FMAs per lane, no memory traffic
    v8f G;
#pragma unroll
    for (int v = 0; v < 8; ++v) {
      float g = 0.0f;
#pragma unroll
      for (int i = 0; i < IN_; ++i) g = fmaf(xn[i] - xc[i], sk[i][v], g);
      G[v] = g;
    }
#pragma unroll
    for (int i = 0; i < IN_; ++i) xc[i] = xn[i];

    // Stage Gh in LDS once; pull its A-fragments (reused all Horner iters)
#pragma unroll
    for (int v = 0; v < 8; ++v) lsA[(v + 8 * hi) * 16 + lo] = G[v];
    __syncthreads();
    v2f ga[4];
#pragma unroll
    for (int kk = 0; kk < 4; ++kk) {
      ga[kk].x = lsA[lo * 16 + (kk * 4 + 2 * hi + 0)];
      ga[kk].y = lsA[lo * 16 + (kk * 4 + 2 * hi + 1)];
    }
    __syncthreads();

    // Horner Taylor deg 8: S = I; for k=8..1: S = I + (Gh/k) @ S
    v8f S;
#pragma unroll
    for (int v = 0; v < 8; ++v) S[v] = ((v + 8 * hi) == lo) ? 1.0f : 0.0f;
#pragma unroll
    for (int k = 8; k >= 1; --k) {
      const float invk = 1.0f / (float)k;
#pragma unroll
      for (int v = 0; v < 8; ++v) lsB[(v + 8 * hi) * 16 + lo] = S[v];
      __syncthreads();
      v8f acc = {};
#pragma unroll
      for (int kk = 0; kk < 4; ++kk) {
        v2f a;
        a.x = ga[kk].x * invk;
        a.y = ga[kk].y * invk;
        v2f b;
        b.x = lsB[(kk * 4 + 2 * hi + 0) * 16 + lo];
        b.y = lsB[(kk * 4 + 2 * hi + 1) * 16 + lo];
        acc = __builtin_amdgcn_wmma_f32_16x16x4_f32(false, a, false, b,
                                                    (short)0, acc, false, false);
      }
      __syncthreads();
#pragma unroll
      for (int v = 0; v < 8; ++v)
        S[v] = acc[v] + (((v + 8 * hi) == lo) ? 1.0f : 0.0f);
    }

    // Square: E = S @ S  ->  exp(G) = exp(G/2)^2
#pragma unroll
    for (int v = 0; v < 8; ++v) {
      lsA[(v + 8 * hi) * 16 + lo] = S[v];
      lsB[(v + 8 * hi) * 16 + lo] = S[v];
    }
    __syncthreads();
    v8f E = {};
    E = mm16_acc(lsA, lsB, E, lo, hi);
    __syncthreads();

    // Ordered chain: P = P @ E
#pragma unroll
    for (int v = 0; v < 8; ++v) {
      lsA[(v + 8 * hi) * 16 + lo] = P[v];
      lsB[(v + 8 * hi) * 16 + lo] = E[v];
    }
    __syncthreads();
    v8f Pn = {};
    Pn = mm16_acc(lsA, lsB, Pn, lo, hi);
    __syncthreads();
    P = Pn;
  }

  // Emit this chunk's 16x16 product tile to workspace
  float* outT = ws + (size_t)wid * 256;
#pragma unroll
  for (int v = 0; v < 8; ++v) outT[(v + 8 * hi) * 16 + lo] = P[v];
}

__global__ __launch_bounds__(32)
void dev_combine_kernel(const float* __restrict__ ws, float* __restrict__ out) {
  __shared__ float lsA[256];
  const int lane = threadIdx.x;
  const int lo = lane & 15;
  const int hi = lane >> 4;
  const int nc = blockIdx.x;                     // 0 .. N*C-1
  const float* base = ws + (size_t)nc * CHUNKS * 256;

  v8f P;
#pragma unroll
  for (int v = 0; v < 8; ++v) P[v] = base[(v + 8 * hi) * 16 + lo];

  for (int qq = 1; qq < CHUNKS; ++qq) {
    const float* tq = base + (size_t)qq * 256;
#pragma unroll
    for (int v = 0; v < 8; ++v) lsA[(v + 8 * hi) * 16 + lo] = P[v];
    __syncthreads();
    v8f acc = {};
#pragma unroll
    for (int kk = 0; kk < 4; ++kk) {
      v2f a, b;
      a.x = lsA[lo * 16 + (kk * 4 + 2 * hi + 0)];
      a.y = lsA[lo * 16 + (kk * 4 + 2 * hi + 1)];
      b.x = tq[(kk * 4 + 2 * hi + 0) * 16 + lo];   // B-frag direct from global
      b.y = tq[(kk * 4 + 2 * hi + 1) * 16 + lo];
      acc = __builtin_amdgcn_wmma_f32_16x16x4_f32(false, a, false, b,
                                                  (short)0, acc, false, false);
    }
    __syncthreads();
    P = acc;
  }

  // Scatter the live 10x10 block to out[(n*C + c), 10, 10]
  float* o = out + (size_t)nc * (M_ * M_);
#pragma unroll
  for (int v = 0; v < 8; ++v) {
    const int r = v + 8 * hi;
    if (r < M_ && lo < M_) o[r * M_ + lo] = P[v];
  }
}

extern "C" void kernel_launch(void* const* d_in, const int* in_sizes, int n_in,
                              void* d_out, int out_size, void* d_ws, size_t ws_size,
                              hipStream_t stream) {
  const float* x = (const float*)d_in[0];   // (64, 2048, 8)
  const float* A = (const float*)d_in[1];   // (2, 8, 10, 10)
  float* out = (float*)d_out;               // (64, 2, 10, 10)
  float* ws  = (float*)d_ws;                // needs N*C*CHUNKS*256*4 = 4 MB

  dim3 grid1(N_ * C_ * CHUNKS);             // 4096 single-wave workgroups
  dim3 grid2(N_ * C_);                      // 128 single-wave workgroups
  dev_chunk_kernel<<<grid1, 32, 0, stream>>>(x, A, ws);
  dev_combine_kernel<<<grid2, 32, 0, stream>>>(ws, out);
}